// RelLayer_32341103739249
// MI455X (gfx1250) — compile-verified
//
#include <hip/hip_runtime.h>
#include <hip/hip_bf16.h>

typedef __attribute__((ext_vector_type(2))) float v2f;
typedef __attribute__((ext_vector_type(8))) float v8f;

#define DIM 64

// ---------------------------------------------------------------------------
// v1 = w_k^T @ e1 ; v2 = w_q^T @ e2     (64 threads, one tiny block)
// ---------------------------------------------------------------------------
__global__ void k_proj_vec(const float* __restrict__ w_q,
                           const float* __restrict__ w_k,
                           const float* __restrict__ e1,
                           const float* __restrict__ e2,
                           float* __restrict__ v1,
                           float* __restrict__ v2) {
    int j = threadIdx.x;                      // 0..63
    float s1 = 0.f, s2 = 0.f;
    #pragma unroll 8
    for (int i = 0; i < DIM; ++i) {
        s1 += w_k[i * DIM + j] * e1[i];
        s2 += w_q[i * DIM + j] * e2[i];
    }
    v1[j] = s1;
    v2[j] = s2;
}

// ---------------------------------------------------------------------------
// h_src[n] = dot(p_u[n], v1) ; h_dst[n] = dot(d_u[n], v2)   (thread per node)
// ---------------------------------------------------------------------------
__global__ void k_node_scores(const float* __restrict__ d_u,
                              const float* __restrict__ p_u,
                              const float* __restrict__ v1,
                              const float* __restrict__ v2,
                              float* __restrict__ h_src,
                              float* __restrict__ h_dst,
                              int n_nodes) {
    int n = blockIdx.x * blockDim.x + threadIdx.x;
    if (n >= n_nodes) return;
    const float4* pr  = (const float4*)(p_u + (size_t)n * DIM);
    const float4* dr  = (const float4*)(d_u + (size_t)n * DIM);
    const float4* v1p = (const float4*)v1;
    const float4* v2p = (const float4*)v2;
    float s1 = 0.f, s2 = 0.f;
    #pragma unroll
    for (int i = 0; i < DIM / 4; ++i) {
        float4 a = pr[i], b = v1p[i];
        s1 += a.x * b.x + a.y * b.y + a.z * b.z + a.w * b.w;
        float4 c = dr[i], d = v2p[i];
        s2 += c.x * d.x + c.y * d.y + c.z * d.z + c.w * d.w;
    }
    h_src[n] = s1;
    h_dst[n] = s2;
}

// ---------------------------------------------------------------------------
// zero fill, float4 granularity (n4 = count of float4 elements)
// ---------------------------------------------------------------------------
__global__ void k_zero4(float4* __restrict__ p, long n4) {
    long i = (long)blockIdx.x * blockDim.x + threadIdx.x;
    if (i < n4) p[i] = make_float4(0.f, 0.f, 0.f, 0.f);
}

// ---------------------------------------------------------------------------
// wv = p_u @ w_v^T via V_WMMA_F32_16X16X4_F32.
// Block = 128 threads = 4 waves. Each wave computes one 16x16 tile:
// rows [blockIdx.x*16, +16), cols [waveId*16, +16).
// A (16x4 f32): lane L(0..15) holds row M=L, K = k0 + {0,1};
//               lane L(16..31) holds row M=L-16, K = k0 + {2,3}.
// B (4x16 f32): lane L(0..15) holds col N=L, K = k0 + {0,1};
//               lane L(16..31) holds col N=L-16, K = k0 + {2,3}.
// C/D (16x16 f32, 8 VGPRs): VGPR r -> M = r + 8*(lane>=16), N = lane&15.
// Row index clamped (not predicated) so EXEC is all-1s for the WMMAs.
// ---------------------------------------------------------------------------
__global__ void k_wv_wmma(const float* __restrict__ p_u,
                          const float* __restrict__ w_v,
                          float* __restrict__ wv,
                          int n_nodes) {
    const int lane    = threadIdx.x & 31;
    const int wave    = threadIdx.x >> 5;      // 0..3 -> column tile
    const int halfSel = lane >> 4;             // 0 or 1
    const int l16     = lane & 15;

    const int row0 = blockIdx.x * 16;
    const int col0 = wave * 16;

    int arow = row0 + l16;
    if (arow >= n_nodes) arow = n_nodes - 1;   // clamp; stores are masked below
    const int bcol = col0 + l16;

    const float* aRow = p_u + (size_t)arow * DIM;          // A row (M fixed, K varies)
    const float* bRow = w_v + (size_t)bcol * DIM;          // B[k][col] == w_v[col][k]

    v8f acc = {};
    #pragma unroll
    for (int k0 = 0; k0 < DIM; k0 += 4) {
        const int ka = k0 + 2 * halfSel;                   // even -> 8B aligned
        v2f a = *(const v2f*)(aRow + ka);
        v2f b = *(const v2f*)(bRow + ka);
        acc = __builtin_amdgcn_wmma_f32_16x16x4_f32(
            /*neg_a=*/false, a, /*neg_b=*/false, b,
            /*c_mod=*/(short)0, acc, /*reuse_a=*/false, /*reuse_b=*/false);
    }

    #pragma unroll
    for (int r = 0; r < 8; ++r) {
        const int m = row0 + r + 8 * halfSel;
        if (m < n_nodes) wv[(size_t)m * DIM + col0 + l16] = acc[r];
    }
}

// ---------------------------------------------------------------------------
// c[e] = h_src[src[e]] + h_dst[dst[e]] ; denom[dst[e]] += c[e]  (atomic)
// ---------------------------------------------------------------------------
__global__ void k_edge_score(const int* __restrict__ src,
                             const int* __restrict__ dst,
                             const float* __restrict__ h_src,
                             const float* __restrict__ h_dst,
                             float* __restrict__ c,
                             float* __restrict__ denom,
                             int n_edges) {
    int e = blockIdx.x * blockDim.x + threadIdx.x;
    if (e >= n_edges) return;
    const int s = src[e];
    const int d = dst[e];
    const float ce = h_src[s] + h_dst[d];
    c[e] = ce;
    atomicAdd(&denom[d], ce);
}

// ---------------------------------------------------------------------------
// out[dst[e]] += wv[src[e]] * (c[e] / denom[dst[e]])
// 16 threads per edge, each owning one float4 chunk of the 64-float row:
//  - row gather becomes 16x global_load_b128 (fully coalesced, 4x fewer
//    vmem instructions than 64x b32)
//  - the redundant src/dst/c/denom loads drop 4x per edge
//  - scatter remains one f32 atomic per element (required by the algorithm)
// ---------------------------------------------------------------------------
__global__ void k_scatter(const int* __restrict__ src,
                          const int* __restrict__ dst,
                          const float* __restrict__ c,
                          const float* __restrict__ denom,
                          const float* __restrict__ wv,
                          float* __restrict__ out,
                          int n_edges) {
    long t = (long)blockIdx.x * blockDim.x + threadIdx.x;
    int e = (int)(t >> 4);                 // edge
    int q = (int)(t & 15);                 // float4 chunk within the row
    if (e >= n_edges) return;
    const int s = src[e];
    const int d = dst[e];
    const float coef = c[e] / denom[d];

    const float4 w = *(const float4*)(wv + (size_t)s * DIM + q * 4);
    float* o = out + (size_t)d * DIM + q * 4;
    atomicAdd(o + 0, w.x * coef);
    atomicAdd(o + 1, w.y * coef);
    atomicAdd(o + 2, w.z * coef);
    atomicAdd(o + 3, w.w * coef);
}

// ---------------------------------------------------------------------------
// launch
// ---------------------------------------------------------------------------
extern "C" void kernel_launch(void* const* d_in, const int* in_sizes, int n_in,
                              void* d_out, int out_size, void* d_ws, size_t ws_size,
                              hipStream_t stream) {
    const float* d_u = (const float*)d_in[0];
    const float* p_u = (const float*)d_in[1];
    const float* w_q = (const float*)d_in[2];
    const float* w_k = (const float*)d_in[3];
    const float* w_v = (const float*)d_in[4];
    const float* e1  = (const float*)d_in[5];
    const float* e2  = (const float*)d_in[6];
    const int*   src = (const int*)d_in[7];
    const int*   dst = (const int*)d_in[8];

    const int n_nodes = in_sizes[0] / DIM;
    const int n_edges = in_sizes[7];

    float* out = (float*)d_out;

    // workspace carve-up (floats); denom padded to multiple of 4 for k_zero4
    const int n_nodes_pad4 = (n_nodes + 3) & ~3;
    float* ws     = (float*)d_ws;
    float* v1     = ws;                          // 64
    float* v2     = v1 + DIM;                    // 64
    float* h_src  = v2 + DIM;                    // n_nodes
    float* h_dst  = h_src + n_nodes;             // n_nodes
    float* denom  = h_dst + n_nodes;             // n_nodes_pad4
    float* c      = denom + n_nodes_pad4;        // n_edges
    float* wv     = c + n_edges;                 // n_nodes * 64

    // 1) fold e1/e2 into projection vectors
    k_proj_vec<<<1, DIM, 0, stream>>>(w_q, w_k, e1, e2, v1, v2);

    // 2) per-node scalar scores
    {
        int blocks = (n_nodes + 255) / 256;
        k_node_scores<<<blocks, 256, 0, stream>>>(d_u, p_u, v1, v2, h_src, h_dst, n_nodes);
    }

    // 3) zero accumulators (denom and the output) every call
    {
        long nd4 = n_nodes_pad4 / 4;
        k_zero4<<<(int)((nd4 + 255) / 256), 256, 0, stream>>>((float4*)denom, nd4);
        long no4 = (long)n_nodes * DIM / 4;
        k_zero4<<<(int)((no4 + 255) / 256), 256, 0, stream>>>((float4*)out, no4);
    }

    // 4) wv = p_u @ w_v^T  (WMMA f32 16x16x4, one wave per 16x16 tile)
    {
        int blocks = (n_nodes + 15) / 16;
        k_wv_wmma<<<blocks, 128, 0, stream>>>(p_u, w_v, wv, n_nodes);
    }

    // 5) edge scores + denominator accumulation
    {
        int blocks = (n_edges + 255) / 256;
        k_edge_score<<<blocks, 256, 0, stream>>>(src, dst, h_src, h_dst, c, denom, n_edges);
    }

    // 6) normalized weighted scatter-add (16 threads / edge, float4 gathers)
    {
        long total = (long)n_edges * 16;
        int blocks = (int)((total + 255) / 256);
        k_scatter<<<blocks, 256, 0, stream>>>(src, dst, c, denom, wv, out, n_edges);
    }
}